// MRConv_19344532701697
// MI455X (gfx1250) — compile-verified
//
#include <hip/hip_runtime.h>
#include <hip/hip_bf16.h>

typedef __attribute__((ext_vector_type(2))) float v2f;
typedef __attribute__((ext_vector_type(4))) float v4f;
typedef __attribute__((ext_vector_type(8))) float v8f;

#define NB 2
#define NC 128
#define ND 48
#define NS (48*48*48)      // 110592 spatial elements
#define HW (48*48)         // 2304

// ---------------------------------------------------------------------------
// Pass 1: per (b,c,d) 48x48 slice, compute H- and W-axis parity minima and
// write xj = max(0, x - min(exclH, exclW)).
// ---------------------------------------------------------------------------
__global__ __launch_bounds__(256) void mrconv_hw_min(const float* __restrict__ x,
                                                     float* __restrict__ xj) {
    __shared__ float s[48 * 49];             // padded rows -> conflict-free cols
    __shared__ float rmin1[96], rmin2[96];   // per (h, parity of w)
    __shared__ float cmin1[96], cmin2[96];   // per (w, parity of h)

    const int tid  = threadIdx.x;
    const size_t base = (size_t)blockIdx.x * HW;   // (b*C+c)*S + d*HW

    #pragma unroll
    for (int i = 0; i < 9; ++i) {
        int idx = tid + i * 256;
        int h = idx / 48, w = idx % 48;
        s[h * 49 + w] = x[base + idx];
    }
    __syncthreads();

    if (tid < 96) {                    // W-axis line minima: row h, parity par
        int h = tid >> 1, par = tid & 1;
        float m1 = 1e30f, m2 = 1e30f;
        for (int w = par; w < 48; w += 2) {
            float v = s[h * 49 + w];
            if (v < m1) { m2 = m1; m1 = v; } else if (v < m2) m2 = v;
        }
        rmin1[tid] = m1; rmin2[tid] = m2;
    } else if (tid < 192) {            // H-axis line minima: col w, parity par
        int t = tid - 96;
        int w = t >> 1, par = t & 1;
        float m1 = 1e30f, m2 = 1e30f;
        for (int h = par; h < 48; h += 2) {
            float v = s[h * 49 + w];
            if (v < m1) { m2 = m1; m1 = v; } else if (v < m2) m2 = v;
        }
        cmin1[t] = m1; cmin2[t] = m2;
    }
    __syncthreads();

    #pragma unroll
    for (int i = 0; i < 9; ++i) {
        int idx = tid + i * 256;
        int h = idx / 48, w = idx % 48;
        float v = s[h * 49 + w];
        int rt = h * 2 + (w & 1);
        int ct = w * 2 + (h & 1);
        float re = (v == rmin1[rt]) ? rmin2[rt] : rmin1[rt];
        float ce = (v == cmin1[ct]) ? cmin2[ct] : cmin1[ct];
        xj[base + idx] = fmaxf(0.0f, v - fminf(re, ce));
    }
}

// ---------------------------------------------------------------------------
// Pass 2: per (b,c,h) 48(d)x48(w) plane, D-axis parity minima; merge into xj.
// ---------------------------------------------------------------------------
__global__ __launch_bounds__(256) void mrconv_d_min(const float* __restrict__ x,
                                                    float* __restrict__ xj) {
    __shared__ float s[48 * 49];
    __shared__ float dmin1[96], dmin2[96];   // per (w, parity of d)

    const int tid = threadIdx.x;
    const int slab = blockIdx.x;             // (b*C+c)*48 + h
    const size_t base = (size_t)(slab / 48) * NS + (size_t)(slab % 48) * 48;

    #pragma unroll
    for (int i = 0; i < 9; ++i) {
        int idx = tid + i * 256;
        int d = idx / 48, w = idx % 48;
        s[d * 49 + w] = x[base + (size_t)d * HW + w];
    }
    __syncthreads();

    if (tid < 96) {
        int w = tid >> 1, par = tid & 1;
        float m1 = 1e30f, m2 = 1e30f;
        for (int d = par; d < 48; d += 2) {
            float v = s[d * 49 + w];
            if (v < m1) { m2 = m1; m1 = v; } else if (v < m2) m2 = v;
        }
        dmin1[tid] = m1; dmin2[tid] = m2;
    }
    __syncthreads();

    #pragma unroll
    for (int i = 0; i < 9; ++i) {
        int idx = tid + i * 256;
        int d = idx / 48, w = idx % 48;
        float v = s[d * 49 + w];
        int t = w * 2 + (d & 1);
        float e = (v == dmin1[t]) ? dmin2[t] : dmin1[t];
        size_t o = base + (size_t)d * HW + w;
        xj[o] = fmaxf(xj[o], v - e);         // xj already >= 0 from pass 1
    }
}

// ---------------------------------------------------------------------------
// Pass 3: fused GEMM  out[o,p] = relu( sum_c W[o,c]*x[c,p] + W[o,c+128]*xj[c,p] + b[o] )
// V_WMMA_F32_16X16X4_F32 (exact fp32 matrix path, wave32). 8 waves/block, each
// wave owns rows [16w,16w+16); W-slice in registers; B tile double-buffered in
// LDS via GLOBAL_LOAD_ASYNC_TO_LDS_B128 (ASYNCcnt); 4 interleaved accumulator
// chains break the WMMA RAW dependency.
// ---------------------------------------------------------------------------
#define NTILES 27

__global__ __launch_bounds__(256) void mrconv_gemm(const float* __restrict__ x,
                                                   const float* __restrict__ xj,
                                                   const float* __restrict__ Wmat,
                                                   const float* __restrict__ bias,
                                                   float* __restrict__ out) {
    __shared__ float Bs[2][256 * 16];        // 2 x 16 KB double buffer

    const int b    = blockIdx.y;
    const int tid  = threadIdx.x;
    const int wave = tid >> 5;
    const int lane = tid & 31;
    const int m    = lane & 15;              // N index (and A row within slice)
    const int kOff = (lane >> 4) * 2;        // A/B K sub-offset per lane half
    const int o0   = wave * 16;

    // Preload this wave's 16x256 weight slice into registers: a[t] covers K=[4t,4t+4)
    v2f a[64];
    #pragma unroll
    for (int t = 0; t < 64; ++t)
        a[t] = *(const v2f*)(Wmat + (size_t)(o0 + m) * 256 + 4 * t + kOff);

    float bv[8];
    #pragma unroll
    for (int j = 0; j < 8; ++j)
        bv[j] = bias[o0 + j + (lane >> 4) * 8];

    const float* xb  = x  + (size_t)b * NC * NS;
    const float* xjb = xj + (size_t)b * NC * NS;

    // Async-stage one 256x16 B tile into LDS buffer `dst` (bypasses VGPRs).
    const int c4 = tid >> 2;                 // 0..63
    const int q  = (tid & 3) * 4;            // 0,4,8,12
    auto stage = [&](int p0, float* dstBase) {
        #pragma unroll
        for (int r = 0; r < 4; ++r) {
            int c = c4 + r * 64;             // 0..255
            const float* src = (c < 128)
                ? (xb  + (size_t)c * NS + p0 + q)
                : (xjb + (size_t)(c - 128) * NS + p0 + q);
            unsigned ldsAddr = (unsigned)(uintptr_t)(dstBase + c * 16 + q);
            asm volatile("global_load_async_to_lds_b128 %0, %1, off"
                         :: "v"(ldsAddr), "v"(src) : "memory");
        }
    };

    const int p0base = blockIdx.x * NTILES * 16;

    // Prologue: stage tile 0 into Bs[0]
    stage(p0base, &Bs[0][0]);
    asm volatile("s_wait_asynccnt 0" ::: "memory");
    __syncthreads();

    for (int tile = 0; tile < NTILES; ++tile) {
        const int buf = tile & 1;
        const int p0  = p0base + tile * 16;

        if (tile + 1 < NTILES)               // overlap next tile's HBM fetch
            stage(p0 + 16, &Bs[1 - buf][0]);

        // 64 WMMAs over K=256, 4 independent accumulator chains (t mod 4)
        v8f acc0 = {}, acc1 = {}, acc2 = {}, acc3 = {};
        const float* Bb = &Bs[buf][0];
        #pragma unroll
        for (int t = 0; t < 64; t += 4) {
            v2f f0, f1, f2, f3;
            f0.x = Bb[(4*t      + kOff    ) * 16 + m];
            f0.y = Bb[(4*t      + kOff + 1) * 16 + m];
            f1.x = Bb[(4*(t+1)  + kOff    ) * 16 + m];
            f1.y = Bb[(4*(t+1)  + kOff + 1) * 16 + m];
            f2.x = Bb[(4*(t+2)  + kOff    ) * 16 + m];
            f2.y = Bb[(4*(t+2)  + kOff + 1) * 16 + m];
            f3.x = Bb[(4*(t+3)  + kOff    ) * 16 + m];
            f3.y = Bb[(4*(t+3)  + kOff + 1) * 16 + m];
            acc0 = __builtin_amdgcn_wmma_f32_16x16x4_f32(false, a[t],   false, f0, (short)0, acc0, false, false);
            acc1 = __builtin_amdgcn_wmma_f32_16x16x4_f32(false, a[t+1], false, f1, (short)0, acc1, false, false);
            acc2 = __builtin_amdgcn_wmma_f32_16x16x4_f32(false, a[t+2], false, f2, (short)0, acc2, false, false);
            acc3 = __builtin_amdgcn_wmma_f32_16x16x4_f32(false, a[t+3], false, f3, (short)0, acc3, false, false);
        }
        v8f acc = (acc0 + acc1) + (acc2 + acc3);

        // My async writes to Bs[1-buf] must land before anyone reads them.
        if (tile + 1 < NTILES)
            asm volatile("s_wait_asynccnt 0" ::: "memory");
        __syncthreads();

        #pragma unroll
        for (int j = 0; j < 8; ++j) {
            const int o = o0 + j + (lane >> 4) * 8;
            float v = acc[j] + bv[j];
            v = v > 0.0f ? v : 0.0f;
            out[((size_t)b * NC + o) * NS + p0 + m] = v;
        }
    }
}

// ---------------------------------------------------------------------------
extern "C" void kernel_launch(void* const* d_in, const int* in_sizes, int n_in,
                              void* d_out, int out_size, void* d_ws, size_t ws_size,
                              hipStream_t stream) {
    const float* x    = (const float*)d_in[0];   // [2,128,48,48,48]
    const float* Wmat = (const float*)d_in[1];   // [128,256]
    const float* bias = (const float*)d_in[2];   // [128]
    float* out = (float*)d_out;                  // [2,128,48,48,48]
    float* xj  = (float*)d_ws;                   // 2*128*110592 floats = 113 MB

    mrconv_hw_min<<<dim3(NB * NC * ND), dim3(256), 0, stream>>>(x, xj);
    mrconv_d_min <<<dim3(NB * NC * ND), dim3(256), 0, stream>>>(x, xj);
    // 110592 / (27 tiles * 16) = 256 blocks per batch, exact tiling (EXEC all-1s)
    mrconv_gemm  <<<dim3(256, NB), dim3(256), 0, stream>>>(x, xj, Wmat, bias, out);
}